// mgfn_loss_51273319580311
// MI455X (gfx1250) — compile-verified
//
#include <hip/hip_runtime.h>
#include <math.h>

typedef float v2f __attribute__((ext_vector_type(2)));
typedef float v8f __attribute__((ext_vector_type(8)));

#define MARGIN_F 200.0f
#define EPS_F 1e-6f

// ---------------------------------------------------------------------------
// Kernel 1: mags[r] = sum_d |feats[r, d]|  for r in [0, 128*1024), D = 1024.
// One wave32 per row; 8x float4 coalesced loads per lane (global_load_b128).
// Bandwidth-bound: 512 MB read @ 23.3 TB/s ~ 22 us.
// ---------------------------------------------------------------------------
__global__ __launch_bounds__(256) void mags_kernel(const float* __restrict__ feats,
                                                   float* __restrict__ mags) {
    const int lane = threadIdx.x & 31;
    const int wave = threadIdx.x >> 5;
    const int row  = blockIdx.x * 8 + wave;        // 16384 blocks * 8 waves = 131072 rows
    const float* p = feats + (size_t)row * 1024;
    float s = 0.0f;
#pragma unroll
    for (int i = 0; i < 8; ++i) {
        float4 v = *(const float4*)(p + i * 128 + lane * 4);
        s += fabsf(v.x) + fabsf(v.y) + fabsf(v.z) + fabsf(v.w);
    }
#pragma unroll
    for (int off = 16; off > 0; off >>= 1) s += __shfl_down(s, off, 32);
    if (lane == 0) mags[row] = s;
}

// ---------------------------------------------------------------------------
// Kernel 2: deterministic BCE-with-logits mean over 131072 elements.
// Single block, LDS tree reduction (no float atomics -> bitwise deterministic).
// ---------------------------------------------------------------------------
__global__ __launch_bounds__(1024) void bce_kernel(const float* __restrict__ scores,
                                                   const float* __restrict__ targets,
                                                   float* __restrict__ loss_cls) {
    __shared__ float red[1024];
    float s = 0.0f;
    for (int i = threadIdx.x; i < 131072; i += 1024) {
        float l = scores[i], t = targets[i];
        s += fmaxf(l, 0.0f) - l * t + log1pf(expf(-fabsf(l)));
    }
    red[threadIdx.x] = s;
    __syncthreads();
    for (int st = 512; st > 0; st >>= 1) {
        if (threadIdx.x < (unsigned)st) red[threadIdx.x] += red[threadIdx.x + st];
        __syncthreads();
    }
    if (threadIdx.x == 0) loss_cls[0] = red[0] / 131072.0f;
}

// ---------------------------------------------------------------------------
// Kernel 3: contrastive term on mags (128 rows x 1024 features), b = 64.
// Gram matrix n_f @ a_f^T (64x64, K=1024) via V_WMMA_F32_16X16X4_F32:
// 16 waves, one 16x16 tile each, 256 WMMA steps over K.
// d2[i,j] = ||n_i||^2 + ||a_j||^2 + K*eps^2 - 2*G[i,j] + 2*eps*(Sn_i - Sa_j).
// Then hinge, first-index argmin, anchored sums, final scalar.
// ---------------------------------------------------------------------------
__global__ __launch_bounds__(512) void contrastive_kernel(const float* __restrict__ mags,
                                                          const float* __restrict__ loss_cls,
                                                          float* __restrict__ out) {
    __shared__ float dmat[64 * 64];        // 16 KB
    __shared__ float ssq[128], ssum[128];  // per-row ||.||^2 and sum
    __shared__ float rvals[512];
    __shared__ int   ridx[512];
    __shared__ float rowsq[128];
    __shared__ int   s_istar, s_jstar;
    __shared__ float s_mind;

    const int tid  = threadIdx.x;
    const int lane = tid & 31;
    const int wave = tid >> 5;

    // ---- Phase A: per-row stats (128 rows / 16 waves = 8 rows each) ----
    for (int r = wave; r < 128; r += 16) {
        const float* p = mags + r * 1024;
        float sq = 0.0f, sm = 0.0f;
        for (int k = lane; k < 1024; k += 32) {
            float v = p[k];
            sq += v * v;
            sm += v;
        }
        for (int off = 16; off > 0; off >>= 1) {
            sq += __shfl_down(sq, off, 32);
            sm += __shfl_down(sm, off, 32);
        }
        if (lane == 0) { ssq[r] = sq; ssum[r] = sm; }
    }
    __syncthreads();

    // ---- Phase B: 64x64 Gram via WMMA f32 16x16x4, K = 1024 ----
    // A-matrix 16x4 f32 layout: lanes 0-15 -> VGPR0:K=0, VGPR1:K=1;
    //                            lanes 16-31 -> VGPR0:K=2, VGPR1:K=3.
    // B-matrix 4x16 layout mirrors it (row K striped across lanes per VGPR).
    const int ti = wave >> 2;        // tile row (0..3)
    const int tj = wave & 3;         // tile col (0..3)
    const int m  = lane & 15;
    const int h  = lane >> 4;        // 0 or 1
    const float* nrow = mags + (ti * 16 + m) * 1024;        // n_f row (M index = m)
    const float* arow = mags + (64 + tj * 16 + m) * 1024;   // a_f row (N index = m)
    v8f c = {0.f, 0.f, 0.f, 0.f, 0.f, 0.f, 0.f, 0.f};
    for (int k0 = 0; k0 < 1024; k0 += 4) {
        v2f av, bv;
        av.x = nrow[k0 + 2 * h];
        av.y = nrow[k0 + 2 * h + 1];
        bv.x = arow[k0 + 2 * h];
        bv.y = arow[k0 + 2 * h + 1];
        c = __builtin_amdgcn_wmma_f32_16x16x4_f32(
                /*neg_a=*/false, av, /*neg_b=*/false, bv,
                /*c_mod=*/(short)0, c, /*reuse_a=*/false, /*reuse_b=*/false);
    }
    const float keps2 = 1024.0f * EPS_F * EPS_F;
#pragma unroll
    for (int r = 0; r < 8; ++r) {
        int i = ti * 16 + r + 8 * h;   // C/D layout: VGPR r -> M=r (lanes 0-15), M=r+8 (16-31)
        int j = tj * 16 + m;
        float g  = c[r];
        float d2 = ssq[i] + ssq[64 + j] + keps2 - 2.0f * g
                 + 2.0f * EPS_F * (ssum[i] - ssum[64 + j]);
        d2 = fmaxf(d2, 0.0f);
        float dist = sqrtf(d2);
        float hng  = fmaxf(MARGIN_F - dist, 0.0f);
        dmat[i * 64 + j] = hng * hng;
    }
    __syncthreads();

    // ---- Phase C: argmin over 4096 values, first-index tie-break ----
    float bestv = 3.402823466e+38f;
    int   besti = 1 << 30;
    for (int q = 0; q < 8; ++q) {
        int   idx = tid * 8 + q;
        float v   = dmat[idx];
        if (v < bestv) { bestv = v; besti = idx; }
    }
    rvals[tid] = bestv;
    ridx[tid]  = besti;
    __syncthreads();
    for (int st = 256; st > 0; st >>= 1) {
        if (tid < st) {
            float v2 = rvals[tid + st];
            int   i2 = ridx[tid + st];
            if (v2 < rvals[tid] || (v2 == rvals[tid] && i2 < ridx[tid])) {
                rvals[tid] = v2;
                ridx[tid]  = i2;
            }
        }
        __syncthreads();
    }
    if (tid == 0) {
        int fi  = ridx[0];
        s_istar = fi >> 6;
        s_jstar = fi & 63;
        s_mind  = rvals[0];
    }
    __syncthreads();

    // ---- Phase D: anchored sums (faithful: square of (x - anchor + eps)) ----
    const int istar = s_istar, jstar = s_jstar;
    for (int r = wave; r < 128; r += 16) {
        const float* p    = mags + r * 1024;
        const float* pref = (r < 64) ? (mags + istar * 1024)
                                     : (mags + (64 + jstar) * 1024);
        float acc = 0.0f;
        for (int k = lane; k < 1024; k += 32) {
            float dv = p[k] - pref[k] + EPS_F;
            acc += dv * dv;
        }
        for (int off = 16; off > 0; off >>= 1) acc += __shfl_down(acc, off, 32);
        if (lane == 0) rowsq[r] = acc;
    }
    __syncthreads();

    if (tid == 0) {
        float sn = 0.0f, sa = 0.0f;
        for (int i = 0; i < 64; ++i)   sn += rowsq[i];
        for (int i = 64; i < 128; ++i) sa += rowsq[i];
        float loss_n   = (sn - rowsq[istar])      / 64.0f;
        float loss_a   = (sa - rowsq[64 + jstar]) / 64.0f;
        float loss_con = 0.001f * s_mind + loss_n + loss_a;
        out[0] = loss_cls[0] + 0.001f * loss_con;
    }
}

extern "C" void kernel_launch(void* const* d_in, const int* in_sizes, int n_in,
                              void* d_out, int out_size, void* d_ws, size_t ws_size,
                              hipStream_t stream) {
    const float* scores  = (const float*)d_in[0];   // (128, 1024)
    const float* feats   = (const float*)d_in[1];   // (128, 1024, 1024)
    const float* targets = (const float*)d_in[2];   // (128, 1024)
    float* out  = (float*)d_out;
    float* mags = (float*)d_ws;              // 131072 floats
    float* lcls = mags + 131072;             // 1 float

    mags_kernel<<<16384, 256, 0, stream>>>(feats, mags);
    bce_kernel<<<1, 1024, 0, stream>>>(scores, targets, lcls);
    contrastive_kernel<<<1, 512, 0, stream>>>(mags, lcls, out);
}